// FTTLinear_60473139528510
// MI455X (gfx1250) — compile-verified
//
#include <hip/hip_runtime.h>
#include <hip/hip_bf16.h>

typedef __attribute__((ext_vector_type(2))) float v2f;
typedef __attribute__((ext_vector_type(4))) float v4f;
typedef __attribute__((ext_vector_type(8))) float v8f;

// One wave computes one 16x16 output tile D[d,b] = sum_j W[c,j,k,d] * X[a,i,j,b]
// for a fixed triple (a,i,c) and k=nt, b-chunk=mt.
//   A (16x4, M=d, K=j)  <- W fragment   (lane = d, VGPRs = K pair per half-wave)
//   B (4x16, K=j, N=b)  <- X fragment   (lane = b)
// D layout: lane = b-column; VGPR r holds d = half*8 + r -> each lane owns 8
// CONSECUTIVE floats of out[..., b*16 + d] -> two global_store_b128 per lane.
template <int M_, int N_, int RA1, int RB0, int P, int RB1, int MT, int NT, int NK>
__global__ __launch_bounds__(256) void tt_core_wmma(const float* __restrict__ x,
                                                    const float* __restrict__ w,
                                                    float* __restrict__ out) {
    const int lane = threadIdx.x & 31;
    const int wave = threadIdx.x >> 5;
    const int lm   = lane & 15;   // A: M=d ; B: N=b (within tile)
    const int half = lane >> 4;   // selects K pair {0,1} vs {2,3}

    int tile = blockIdx.x * 8 + wave;
    const int nt = tile % NT;  tile /= NT;   // k
    const int mt = tile % MT;  tile /= MT;   // 16-row chunk of b
    const int c  = tile % RB0; tile /= RB0;
    const int i  = tile % M_;
    const int a  = tile / M_;

    // x[a,i,j,b] : index ((a*M_+i)*N_ + j)*RA1 + b      (B fragment, lane=b)
    const long xbase = (long)(a * M_ + i) * N_ * RA1 + mt * 16 + lm;
    // w[c,j,k,d] : index ((c*N_+j)*P + k)*RB1 + d       (A fragment, lane=d)
    const long wbase = ((long)c * N_ * P + nt) * RB1 + lm;

    v8f acc = {0.f, 0.f, 0.f, 0.f, 0.f, 0.f, 0.f, 0.f};
#pragma unroll
    for (int kk = 0; kk < NK; ++kk) {
        // 32-bit 16x4 operand layout: lanes 0-15 hold K=j0,j0+1; lanes 16-31 K=j0+2,j0+3
        const int j0 = kk * 4 + half * 2;
        const int j1 = j0 + 1;
        v2f A, B;
        if constexpr (NK * 4 == N_) {
            A.x = w[wbase + (long)j0 * (P * RB1)];
            A.y = w[wbase + (long)j1 * (P * RB1)];
            B.x = x[xbase + (long)j0 * RA1];
            B.y = x[xbase + (long)j1 * RA1];
        } else {
            // branch-free K padding (keeps EXEC untouched): clamp index, mask value
            const int   jc0 = (j0 < N_) ? j0 : 0;
            const int   jc1 = (j1 < N_) ? j1 : 0;
            const float m0  = (j0 < N_) ? 1.f : 0.f;
            const float m1  = (j1 < N_) ? 1.f : 0.f;
            A.x = w[wbase + (long)jc0 * (P * RB1)] * m0;
            A.y = w[wbase + (long)jc1 * (P * RB1)] * m1;
            B.x = x[xbase + (long)jc0 * RA1] * m0;
            B.y = x[xbase + (long)jc1 * RA1] * m1;
        }
        // D = A(16x4) * B(4x16) + C, fp32 accumulate
        acc = __builtin_amdgcn_wmma_f32_16x16x4_f32(false, A, false, B,
                                                    (short)0, acc, false, false);
    }

    // out[(a*RB0+c), i, k, b*RB1+d] ; this wave's tile is a contiguous 1KB block.
    // lane owns b = mt*16+lm, d = half*8 + 0..7 -> 8 consecutive floats (32B aligned)
    const long R1    = (long)RA1 * RB1;
    const long obase = (((long)(a * RB0 + c) * M_ + i) * P + nt) * R1;
    float* p = out + obase + (long)(mt * 16 + lm) * RB1 + half * 8;
    v4f lo, hi;
    lo.x = acc[0]; lo.y = acc[1]; lo.z = acc[2]; lo.w = acc[3];
    hi.x = acc[4]; hi.y = acc[5]; hi.z = acc[6]; hi.w = acc[7];
    __builtin_nontemporal_store(lo, (v4f*)p);        // streaming: output never re-read
    __builtin_nontemporal_store(hi, (v4f*)(p + 4));
}

// Core 4: ra1 == 1 -> degenerate (M=1); 2048 scalar dot products of length 8.
// out[(a*16+c)*2 + i] = sum_j x4[(a*2+i)*8 + j] * w4[c*8 + j]
__global__ __launch_bounds__(256) void tt_core4_scalar(const float* __restrict__ x,
                                                       const float* __restrict__ w,
                                                       float* __restrict__ out) {
    const int t = blockIdx.x * 256 + threadIdx.x; // 0..2047
    const int i = t & 1;
    const int c = (t >> 1) & 15;
    const int a = t >> 5;
    float s = 0.f;
#pragma unroll
    for (int j = 0; j < 8; ++j)
        s += x[(a * 2 + i) * 8 + j] * w[c * 8 + j];
    out[t] = s;
}

extern "C" void kernel_launch(void* const* d_in, const int* in_sizes, int n_in,
                              void* d_out, int out_size, void* d_ws, size_t ws_size,
                              hipStream_t stream) {
    const float* x0 = (const float*)d_in[0];
    const float* x1 = (const float*)d_in[1];
    const float* x2 = (const float*)d_in[2];
    const float* x3 = (const float*)d_in[3];
    const float* x4 = (const float*)d_in[4];
    const float* w0 = (const float*)d_in[5];
    const float* w1 = (const float*)d_in[6];
    const float* w2 = (const float*)d_in[7];
    const float* w3 = (const float*)d_in[8];
    const float* w4 = (const float*)d_in[9];

    float* out = (float*)d_out;
    float* o0 = out;                 // [1,8,1,1024]      =      8192
    float* o1 = o0 + 8192;           // [1024,8,2,1024]   =  16777216
    float* o2 = o1 + 16777216;       // [1024,8,5,1024]   =  41943040
    float* o3 = o2 + 41943040;       // [1024,4,1,1024]   =   4194304
    float* o4 = o3 + 4194304;        // [1024,2,1,1]      =      2048

    // core0: x(1,8,3,64)  w(1,3,1,16)   tiles = 1*8*1*4*1   =     32 ->    4 blocks
    tt_core_wmma<8, 3, 64, 1, 1, 16, 4, 1, 1><<<4, 256, 0, stream>>>(x0, w0, o0);
    // core1: x(64,8,4,64) w(16,4,2,16)  tiles = 64*8*16*4*2 =  65536 -> 8192 blocks
    tt_core_wmma<8, 4, 64, 16, 2, 16, 4, 2, 1><<<8192, 256, 0, stream>>>(x1, w1, o1);
    // core2: x(64,8,8,64) w(16,8,5,16)  tiles = 64*8*16*4*5 = 163840 -> 20480 blocks
    tt_core_wmma<8, 8, 64, 16, 5, 16, 4, 5, 2><<<20480, 256, 0, stream>>>(x2, w2, o2);
    // core3: x(64,4,4,64) w(16,4,1,16)  tiles = 64*4*16*4*1 =  16384 -> 2048 blocks
    tt_core_wmma<4, 4, 64, 16, 1, 16, 4, 1, 1><<<2048, 256, 0, stream>>>(x3, w3, o3);
    // core4: scalar, 2048 outputs
    tt_core4_scalar<<<8, 256, 0, stream>>>(x4, w4, o4);
}